// ImgOnlyOnnxRelative_13322988552663
// MI455X (gfx1250) — compile-verified
//
#include <hip/hip_runtime.h>
#include <cstdint>
#include <cstddef>

// Problem geometry (fixed by the reference).
#define H_DIM 640
#define W_DIM 480
#define NPIX (H_DIM * W_DIM)          // 307200 pixels
#define MASK_DWORDS (NPIX / 32)       // 9600 dwords per polarity bitmask
#define SLAB_DWORDS (2 * MASK_DWORDS) // 19200 dwords: [neg | pos]
#define SLAB_BYTES (SLAB_DWORDS * 4)  // 76800 bytes = 75 KB (fits CDNA5 320KB LDS)
#define MAX_SLABS 256
#define DELTA 15.0f

typedef int      v4i __attribute__((ext_vector_type(4)));
typedef unsigned v4u __attribute__((ext_vector_type(4)));

// ---------------------------------------------------------------------------
// Kernel 1: per-workgroup privatized scatter into LDS bitmasks (ds_or_b32),
// event streams read with non-temporal hints (201 MB read-once > 192 MB L2),
// then async LDS -> global slab dump (global_store_async_from_lds_b128).
// ---------------------------------------------------------------------------
__global__ void ev_scatter(const int* __restrict__ ex,
                           const int* __restrict__ ey,
                           const int* __restrict__ ep,
                           unsigned* __restrict__ slabs,
                           int nEvents) {
  extern __shared__ unsigned lmask[];  // SLAB_DWORDS dwords = 75 KB
  const int tid  = threadIdx.x;
  const int nthr = blockDim.x;

  // Zero this workgroup's private bitmasks.
  for (int d = tid; d < SLAB_DWORDS; d += nthr) lmask[d] = 0u;
  __syncthreads();

  // Grid-stride over events, 4 per thread per stream (b128 coalesced NT loads).
  const long long gtid    = (long long)blockIdx.x * nthr + tid;
  const long long gstride = (long long)gridDim.x * nthr;
  const int nQuads = nEvents >> 2;
  const v4i* __restrict__ ex4 = (const v4i*)ex;
  const v4i* __restrict__ ey4 = (const v4i*)ey;
  const v4i* __restrict__ ep4 = (const v4i*)ep;

  for (long long i = gtid; i < nQuads; i += gstride) {
    const v4i x = __builtin_nontemporal_load(&ex4[i]);
    const v4i y = __builtin_nontemporal_load(&ey4[i]);
    const v4i p = __builtin_nontemporal_load(&ep4[i]);

    unsigned pix;
    pix = (unsigned)(x.x * W_DIM + y.x);
    atomicOr(&lmask[(unsigned)p.x * MASK_DWORDS + (pix >> 5)], 1u << (pix & 31));
    pix = (unsigned)(x.y * W_DIM + y.y);
    atomicOr(&lmask[(unsigned)p.y * MASK_DWORDS + (pix >> 5)], 1u << (pix & 31));
    pix = (unsigned)(x.z * W_DIM + y.z);
    atomicOr(&lmask[(unsigned)p.z * MASK_DWORDS + (pix >> 5)], 1u << (pix & 31));
    pix = (unsigned)(x.w * W_DIM + y.w);
    atomicOr(&lmask[(unsigned)p.w * MASK_DWORDS + (pix >> 5)], 1u << (pix & 31));
  }

  // Scalar tail (nEvents not divisible by 4); one thread, negligible.
  if (gtid == 0) {
    for (int i = nQuads << 2; i < nEvents; ++i) {
      unsigned pix = (unsigned)(ex[i] * W_DIM + ey[i]);
      atomicOr(&lmask[(unsigned)ep[i] * MASK_DWORDS + (pix >> 5)], 1u << (pix & 31));
    }
  }
  __syncthreads();

  // Async dump: LDS bitmask -> this workgroup's global slab, 16B per lane per
  // op, contiguous 512B per wave. Bypasses VGPRs entirely (ASYNCcnt-tracked).
  unsigned* __restrict__ slab = slabs + (size_t)blockIdx.x * SLAB_DWORDS;
  for (int d = tid * 4; d < SLAB_DWORDS; d += nthr * 4) {
    // Low 32 bits of the generic address of an LDS object = wave-relative LDS
    // byte offset (shared aperture layout).
    unsigned ldsOff = (unsigned)(unsigned long long)(const void*)&lmask[d];
    unsigned long long ga = (unsigned long long)(slab + d);
    asm volatile("global_store_async_from_lds_b128 %0, %1, off"
                 :
                 : "v"(ga), "v"(ldsOff)
                 : "memory");
  }
  asm volatile("s_wait_asynccnt 0x0" ::: "memory");
}

// ---------------------------------------------------------------------------
// Kernel 2: OR-reduce the G slabs into the final bitmask (uint4 rows,
// coalesced b128 loads, L2-hot working set of at most ~20 MB).
// ---------------------------------------------------------------------------
__global__ void mask_reduce(const unsigned* __restrict__ slabs,
                            unsigned* __restrict__ fmask, int G) {
  int d4 = blockIdx.x * blockDim.x + threadIdx.x;
  if (d4 >= SLAB_DWORDS / 4) return;
  const v4u* __restrict__ slabs4 = (const v4u*)slabs;
  v4u acc = {0u, 0u, 0u, 0u};
  for (int s = 0; s < G; ++s) {
    v4u r = slabs4[(size_t)s * (SLAB_DWORDS / 4) + d4];
    acc |= r;
  }
  ((v4u*)fmask)[d4] = acc;
}

// ---------------------------------------------------------------------------
// Kernel 3: apply masks to the picture, 4 pixels per thread (float4 I/O).
// Pixels 4t..4t+3 always share one mask dword (4-aligned bit groups).
// ---------------------------------------------------------------------------
__global__ void mask_apply(const float* __restrict__ pic,
                           const unsigned* __restrict__ fmask,
                           float* __restrict__ out) {
  int t = blockIdx.x * blockDim.x + threadIdx.x;
  if (t >= NPIX / 4) return;
  const unsigned shift = (unsigned)(t & 7) * 4u;
  const unsigned nm = (fmask[t >> 3] >> shift) & 0xFu;
  const unsigned pm = (fmask[MASK_DWORDS + (t >> 3)] >> shift) & 0xFu;

  const float4 v = ((const float4*)pic)[t];
  float4 r;
  r.x = v.x + DELTA * ((float)((pm >> 0) & 1u) - (float)((nm >> 0) & 1u));
  r.y = v.y + DELTA * ((float)((pm >> 1) & 1u) - (float)((nm >> 1) & 1u));
  r.z = v.z + DELTA * ((float)((pm >> 2) & 1u) - (float)((nm >> 2) & 1u));
  r.w = v.w + DELTA * ((float)((pm >> 3) & 1u) - (float)((nm >> 3) & 1u));
  ((float4*)out)[t] = r;
}

// ---------------------------------------------------------------------------
extern "C" void kernel_launch(void* const* d_in, const int* in_sizes, int n_in,
                              void* d_out, int out_size, void* d_ws, size_t ws_size,
                              hipStream_t stream) {
  const int*   ex  = (const int*)d_in[0];
  const int*   ey  = (const int*)d_in[1];
  const int*   ep  = (const int*)d_in[2];
  const float* pic = (const float*)d_in[3];
  float*       out = (float*)d_out;
  const int nEvents = in_sizes[0];

  // Workspace: G slabs + 1 final mask, all SLAB_BYTES each.
  int G = (int)(ws_size / (size_t)SLAB_BYTES) - 1;
  if (G > MAX_SLABS) G = MAX_SLABS;
  if (G < 1) G = 1;

  unsigned* slabs = (unsigned*)d_ws;
  unsigned* fmask = slabs + (size_t)G * SLAB_DWORDS;

  // 75 KB dynamic LDS per workgroup (> 64 KB default cap -> raise attribute).
  (void)hipFuncSetAttribute(reinterpret_cast<const void*>(&ev_scatter),
                            hipFuncAttributeMaxDynamicSharedMemorySize,
                            SLAB_BYTES);

  ev_scatter<<<G, 1024, SLAB_BYTES, stream>>>(ex, ey, ep, slabs, nEvents);
  mask_reduce<<<(SLAB_DWORDS / 4 + 255) / 256, 256, 0, stream>>>(slabs, fmask, G);
  mask_apply<<<(NPIX / 4 + 255) / 256, 256, 0, stream>>>(pic, fmask, out);
}